// EncodingNetwork_60627758350926
// MI455X (gfx1250) — compile-verified
//
#include <hip/hip_runtime.h>

typedef float v2f __attribute__((ext_vector_type(2)));
typedef float v8f __attribute__((ext_vector_type(8)));

#define EMBD   1024
#define NWAY   20
#define NSUP   10
#define MPAD   32          // 20 rows padded to 2 WMMA M-tiles

#define WMMA_F32(a, b, c) \
    __builtin_amdgcn_wmma_f32_16x16x4_f32(false, (a), false, (b), (short)0, (c), false, false)

// ---------------------------------------------------------------------------
// Kernel A: per-class mean of x_support -> xm_pad [32][1024], rows 20..31 = 0
// ---------------------------------------------------------------------------
__global__ void class_mean_kernel(const float* __restrict__ x,
                                  float* __restrict__ xm) {
    int idx = blockIdx.x * blockDim.x + threadIdx.x;   // 0 .. 32*1024
    int c = idx >> 10;
    int d = idx & (EMBD - 1);
    float s = 0.0f;
    if (c < NWAY) {
        const float* row = x + (size_t)(c * NSUP) * EMBD + d;
#pragma unroll
        for (int i = 0; i < NSUP; ++i) s += row[(size_t)i * EMBD];
        s *= (1.0f / NSUP);
    }
    xm[idx] = s;
}

// ---------------------------------------------------------------------------
// WMMA GEMM:  C[32][N] = A[32][K] @ B^T  (+ per-column bias as C init)
//   A row-major, lda == K.  B row j (= output column j) at B + j*ldb, length K.
//   One wave per 16-col N-tile; the wave carries BOTH 16-row M-tiles so each
//   weight element is loaded exactly once (weights dominate DRAM traffic).
//   Software-pipelined: next step's fragments are loaded before the current
//   WMMAs so demand loads overlap matrix issue; last chunk peeled to keep
//   every load strictly in-bounds.
//
// f32 16x16x4 fragment layout (ISA 7.12.2):
//   A 16x4:  lane m / m+16 hold row m; VGPR0,1 = K {k,k+1} (lanes 0-15) or
//            {k+2,k+3} (lanes 16-31)
//   B 4x16:  lane n / n+16 hold col n; same K-pair split
//   C/D:     VGPR r -> row r + 8*(lane>=16), col = lane&15
// ---------------------------------------------------------------------------
__global__ void wmma_gemm_kernel(const float* __restrict__ A,
                                 const float* __restrict__ B,
                                 const float* __restrict__ bias,
                                 float* __restrict__ C,
                                 int K, int ldb, int ldc) {
    const int wave = threadIdx.x >> 5;
    const int lane = threadIdx.x & 31;
    const int nt   = blockIdx.x * (blockDim.x >> 5) + wave;
    const int jn   = nt * 16 + (lane & 15); // output column == B row index
    const int kk0  = 2 * (lane >> 4);       // K pair {k,k+1} vs {k+2,k+3}
    const int mrow = lane & 15;

    const float bv = bias[jn];
    v8f acc0 = { bv, bv, bv, bv, bv, bv, bv, bv };
    v8f acc1 = acc0;

    const float* __restrict__ Ar0 = A + (size_t)mrow * K;
    const float* __restrict__ Ar1 = A + (size_t)(16 + mrow) * K;
    const float* __restrict__ Br  = B + (size_t)jn * ldb;

    // pipeline prologue: fragments for k-step 0
    v2f a0 = *(const v2f*)(Ar0 + kk0);
    v2f a1 = *(const v2f*)(Ar1 + kk0);
    v2f b  = *(const v2f*)(Br  + kk0);

    // main chunks: compute 8 steps, always pre-loading the following step
    for (int k = 0; k < K - 32; k += 32) {
        // Uniform stream-prefetch ~4KB ahead on this lane's weight row.
        __builtin_prefetch(Br + k + 1024, 0, 1);
#pragma unroll
        for (int k4 = 4; k4 <= 32; k4 += 4) {
            const int kn = k + k4 + kk0;
            v2f na0 = *(const v2f*)(Ar0 + kn);
            v2f na1 = *(const v2f*)(Ar1 + kn);
            v2f nb  = *(const v2f*)(Br  + kn);
            acc0 = WMMA_F32(a0, b, acc0);
            acc1 = WMMA_F32(a1, b, acc1);
            a0 = na0; a1 = na1; b = nb;
        }
    }
    // last chunk: 7 pipelined steps + drain (no loads past K)
#pragma unroll
    for (int k4 = 4; k4 <= 28; k4 += 4) {
        const int kn = (K - 32) + k4 + kk0;
        v2f na0 = *(const v2f*)(Ar0 + kn);
        v2f na1 = *(const v2f*)(Ar1 + kn);
        v2f nb  = *(const v2f*)(Br  + kn);
        acc0 = WMMA_F32(a0, b, acc0);
        acc1 = WMMA_F32(a1, b, acc1);
        a0 = na0; a1 = na1; b = nb;
    }
    acc0 = WMMA_F32(a0, b, acc0);
    acc1 = WMMA_F32(a1, b, acc1);

#pragma unroll
    for (int r = 0; r < 8; ++r) {
        const int row = r + 4 * kk0;       // r + 8*half
        C[(size_t)row * ldc + jn]        = acc0[r];
        C[(size_t)(16 + row) * ldc + jn] = acc1[r];
    }
}

// ---------------------------------------------------------------------------
// Kernel C: mean over the 20 class rows of encm -> mean_enc [1024]
// ---------------------------------------------------------------------------
__global__ void colmean_kernel(const float* __restrict__ encm,
                               float* __restrict__ me) {
    int d = blockIdx.x * blockDim.x + threadIdx.x;     // 0..1023
    float s = 0.0f;
#pragma unroll
    for (int c = 0; c < NWAY; ++c) s += encm[(size_t)c * EMBD + d];
    me[d] = s * (1.0f / NWAY);
}

// ---------------------------------------------------------------------------
// Kernel D: v[j] = mean_enc . Wr[j,:] + b_rel[j],  Wr[j,k] = W_rel[j*2048+1024+k]
// One wave per output row, full-bandwidth streaming read of the 8 MB Wr block.
// ---------------------------------------------------------------------------
__global__ void gemv_kernel(const float* __restrict__ me,
                            const float* __restrict__ Wrel,
                            const float* __restrict__ brel,
                            float* __restrict__ v) {
    const int wave = threadIdx.x >> 5;
    const int lane = threadIdx.x & 31;
    const int j    = blockIdx.x * (blockDim.x >> 5) + wave;
    const float* __restrict__ row = Wrel + (size_t)j * (2 * EMBD) + EMBD;
    float s = 0.0f;
#pragma unroll 8
    for (int k = lane; k < EMBD; k += 32) s += me[k] * row[k];
#pragma unroll
    for (int off = 16; off > 0; off >>= 1) s += __shfl_xor(s, off, 32);
    if (lane == 0) v[j] = s + brel[j];
}

// ---------------------------------------------------------------------------
// Kernel E: out[c,d] = g[c,d] * pc[c,1024+d] + pc[c,d]
// ---------------------------------------------------------------------------
__global__ void finalize_kernel(const float* __restrict__ pc,
                                const float* __restrict__ g,
                                float* __restrict__ out) {
    int idx = blockIdx.x * blockDim.x + threadIdx.x;   // 0 .. 20*1024
    int c = idx >> 10;
    int d = idx & (EMBD - 1);
    const float* row = pc + (size_t)c * (2 * EMBD);
    out[idx] = fmaf(g[idx], row[EMBD + d], row[d]);
}

// ---------------------------------------------------------------------------
extern "C" void kernel_launch(void* const* d_in, const int* in_sizes, int n_in,
                              void* d_out, int out_size, void* d_ws, size_t ws_size,
                              hipStream_t stream) {
    (void)in_sizes; (void)n_in; (void)out_size; (void)ws_size;

    const float* x_support = (const float*)d_in[0];   // [200,1024]
    const float* W_enc     = (const float*)d_in[1];   // [1024,1024]
    const float* b_enc     = (const float*)d_in[2];   // [1024]
    const float* W_rel     = (const float*)d_in[3];   // [2048,2048]
    const float* b_rel     = (const float*)d_in[4];   // [2048]
    const float* gvec      = (const float*)d_in[5];   // [20,1024]
    float*       out       = (float*)d_out;           // [20,1024]

    float* ws   = (float*)d_ws;
    float* xm   = ws;                       // [32][1024]
    float* encm = xm   + MPAD * EMBD;       // [32][1024]
    float* me   = encm + MPAD * EMBD;       // [1024]
    float* v    = me   + EMBD;              // [2048]
    float* pc   = v    + 2 * EMBD;          // [32][2048]

    // A: class means of x (padded/zeroed to 32 rows)
    class_mean_kernel<<<(MPAD * EMBD) / 256, 256, 0, stream>>>(x_support, xm);

    // B: encm = xm @ W_enc^T + b_enc      (64 N-tiles, 8 waves/block)
    wmma_gemm_kernel<<<8, 256, 0, stream>>>(xm, W_enc, b_enc, encm,
                                            EMBD, EMBD, EMBD);

    // C: mean_enc
    colmean_kernel<<<EMBD / 256, 256, 0, stream>>>(encm, me);

    // D: v = mean_enc @ Wr^T + b_rel      (2048 rows, one wave each)
    gemv_kernel<<<2 * EMBD / 8, 256, 0, stream>>>(me, W_rel, b_rel, v);

    // E: pc = encm @ Wl^T + v             (128 N-tiles; Wl = W_rel[:, :1024])
    wmma_gemm_kernel<<<16, 256, 0, stream>>>(encm, W_rel, v, pc,
                                             EMBD, 2 * EMBD, 2 * EMBD);

    // F: out = g * stds + means
    finalize_kernel<<<(NWAY * EMBD) / 256, 256, 0, stream>>>(pc, gvec, out);
}